// Transform3D_27204322853442
// MI455X (gfx1250) — compile-verified
//
#include <hip/hip_runtime.h>
#include <math.h>

typedef float v2f __attribute__((ext_vector_type(2)));
typedef float v8f __attribute__((ext_vector_type(8)));

// ---------------------------------------------------------------------------
// Stage 1: one wave32 computes mu_rho = im_feat @ pos_w.T + pos_b via
// V_WMMA_F32_16X16X4_F32 (A: 16x4 f32, B: 4x16 f32, C/D: 16x16 f32),
// then builds the 3x3 rotation matrices R[b] = rot_y(mu_b) @ rot_x(rho_b).
//
// A layout (ISA 7.12.2, 32-bit A 16x4): lane l -> M = l&15; VGPR0/1 hold
// K = 2*(l>>4) + {0,1}.   B layout (mirrors C/D striping, 4 rows over 2
// VGPRs): lane l -> N = l&15; VGPR0/1 hold K = 2*(l>>4) + {0,1}.
// D layout: VGPR v, lanes 0-15 -> (M=v, N=lane); so lane 0 holds mu for all
// 8 batches, lane 1 holds rho.
// ---------------------------------------------------------------------------
__global__ __launch_bounds__(32)
void rot_setup_kernel(const float* __restrict__ im_feat,  // [8,512]
                      const float* __restrict__ pos_w,    // [2,512]
                      const float* __restrict__ pos_b,    // [2]
                      float* __restrict__ Rout)           // [8,9]
{
    __shared__ float sh[16];
    const int lane = threadIdx.x;
    const int M    = lane & 15;
    const int half = lane >> 4;

    // Branchless padding of the 8x512 / 2x512 operands to the 16-wide tile.
    const float amask = (M < 8) ? 1.0f : 0.0f;
    const float bmask = (M < 2) ? 1.0f : 0.0f;
    const int   Mr    = (M < 8) ? M : 0;
    const int   Nr    = (M < 2) ? M : 0;

    const float* arow = im_feat + Mr * 512 + 2 * half;
    const float* brow = pos_w   + Nr * 512 + 2 * half;

    v8f acc = {};
    for (int k0 = 0; k0 < 512; k0 += 4) {
        v2f a, b;
        a.x = arow[k0 + 0] * amask;
        a.y = arow[k0 + 1] * amask;
        b.x = brow[k0 + 0] * bmask;
        b.y = brow[k0 + 1] * bmask;
        // 8 args: (neg_a, A, neg_b, B, c_mod, C, reuse_a, reuse_b)
        acc = __builtin_amdgcn_wmma_f32_16x16x4_f32(
            false, a, false, b, (short)0, acc, false, false);
    }

    // Lane N (0 = mu, 1 = rho) holds mu_rho[M=v][N] in acc[v], v = batch.
    if (half == 0 && M < 2) {
        const float pb = pos_b[M];
        #pragma unroll
        for (int v = 0; v < 8; ++v) sh[v * 2 + M] = acc[v] + pb;
    }
    __syncthreads();

    if (lane < 8) {
        const float mu  = sh[lane * 2 + 0];
        const float rho = sh[lane * 2 + 1];
        const float cm = cosf(mu),  sm = sinf(mu);
        const float cr = cosf(rho), sr = sinf(rho);
        float* Rb = Rout + lane * 9;
        // R = rot_y @ rot_x
        Rb[0] = cm;        Rb[1] = sm * sr;   Rb[2] = -sm * cr;
        Rb[3] = 0.0f;      Rb[4] = cr;        Rb[5] = sr;
        Rb[6] = sm;        Rb[7] = -cm * sr;  Rb[8] = cm * cr;
    }
}

// ---------------------------------------------------------------------------
// Stage 2: rotated trilinear grid sample. One thread per (b,z,y,x), all 4
// channels. blockIdx.y = batch so the 32 MB per-batch source slice stays
// L2-resident; output written with non-temporal stores to keep it that way.
// ---------------------------------------------------------------------------
__global__ __launch_bounds__(256)
void grid_sample_kernel(const float* __restrict__ x,   // [8,4,128,128,128]
                        const float* __restrict__ R,   // [8,9]
                        float* __restrict__ out)       // [8,4,128,128,128]
{
    constexpr int DIM  = 128;
    constexpr int C    = 4;
    constexpr int NVOX = DIM * DIM * DIM;

    const int b  = blockIdx.y;
    const int t  = blockIdx.x * 256 + threadIdx.x;
    const int xw = t & (DIM - 1);
    const int yh = (t >> 7) & (DIM - 1);
    const int zd = t >> 14;

    const float* Rb = R + b * 9;
    const float r00 = Rb[0], r01 = Rb[1], r02 = Rb[2];
    const float r10 = Rb[3], r11 = Rb[4], r12 = Rb[5];
    const float r20 = Rb[6], r21 = Rb[7], r22 = Rb[8];

    const float step = 2.0f / (float)(DIM - 1);
    const float X = fmaf((float)xw, step, -1.0f);
    const float Y = fmaf((float)yh, step, -1.0f);
    const float Z = fmaf((float)zd, step, -1.0f);

    const float gx = fmaf(r00, X, fmaf(r01, Y, r02 * Z));
    const float gy = fmaf(r10, X, fmaf(r11, Y, r12 * Z));
    const float gz = fmaf(r20, X, fmaf(r21, Y, r22 * Z));

    // align_corners=False unnormalization
    const float ix = ((gx + 1.0f) * (float)DIM - 1.0f) * 0.5f;
    const float iy = ((gy + 1.0f) * (float)DIM - 1.0f) * 0.5f;
    const float iz = ((gz + 1.0f) * (float)DIM - 1.0f) * 0.5f;

    const float ix0f = floorf(ix), iy0f = floorf(iy), iz0f = floorf(iz);
    const float fx = ix - ix0f, fy = iy - iy0f, fz = iz - iz0f;
    const int ix0 = (int)ix0f, iy0 = (int)iy0f, iz0 = (int)iz0f;

    const float wx[2] = {1.0f - fx, fx};
    const float wy[2] = {1.0f - fy, fy};
    const float wz[2] = {1.0f - fz, fz};

    int   idx[8];
    float wgt[8];
    #pragma unroll
    for (int dz = 0; dz < 2; ++dz) {
        #pragma unroll
        for (int dy = 0; dy < 2; ++dy) {
            #pragma unroll
            for (int dx = 0; dx < 2; ++dx) {
                const int xi = ix0 + dx, yi = iy0 + dy, zi = iz0 + dz;
                const bool valid = ((unsigned)xi < (unsigned)DIM) &
                                   ((unsigned)yi < (unsigned)DIM) &
                                   ((unsigned)zi < (unsigned)DIM);
                const int xc = min(max(xi, 0), DIM - 1);
                const int yc = min(max(yi, 0), DIM - 1);
                const int zc = min(max(zi, 0), DIM - 1);
                const int k = (dz << 2) | (dy << 1) | dx;
                idx[k] = (zc * DIM + yc) * DIM + xc;
                wgt[k] = valid ? (wx[dx] * wy[dy] * wz[dz]) : 0.0f;
            }
        }
    }

    const float* xb = x + (size_t)b * C * NVOX;
    float a0 = 0.0f, a1 = 0.0f, a2 = 0.0f, a3 = 0.0f;
    #pragma unroll
    for (int k = 0; k < 8; ++k) {
        const float w = wgt[k];
        const int   id = idx[k];
        a0 = fmaf(xb[id + 0 * NVOX], w, a0);
        a1 = fmaf(xb[id + 1 * NVOX], w, a1);
        a2 = fmaf(xb[id + 2 * NVOX], w, a2);
        a3 = fmaf(xb[id + 3 * NVOX], w, a3);
    }

    float* ob = out + (size_t)b * C * NVOX + t;
    __builtin_nontemporal_store(a0, ob + 0 * NVOX);
    __builtin_nontemporal_store(a1, ob + 1 * NVOX);
    __builtin_nontemporal_store(a2, ob + 2 * NVOX);
    __builtin_nontemporal_store(a3, ob + 3 * NVOX);
}

extern "C" void kernel_launch(void* const* d_in, const int* in_sizes, int n_in,
                              void* d_out, int out_size, void* d_ws, size_t ws_size,
                              hipStream_t stream) {
    const float* x       = (const float*)d_in[0];  // [8,4,128,128,128]
    const float* im_feat = (const float*)d_in[1];  // [8,512]
    const float* pos_w   = (const float*)d_in[2];  // [2,512]
    const float* pos_b   = (const float*)d_in[3];  // [2]
    float*       out     = (float*)d_out;
    float*       Rws     = (float*)d_ws;           // 8*9 floats

    rot_setup_kernel<<<1, 32, 0, stream>>>(im_feat, pos_w, pos_b, Rws);

    constexpr int NVOX = 128 * 128 * 128;
    dim3 grid(NVOX / 256, 8, 1);
    grid_sample_kernel<<<grid, 256, 0, stream>>>(x, Rws, out);
}